// centernet_Loss_53738630807912
// MI455X (gfx1250) — compile-verified
//
#include <hip/hip_runtime.h>
#include <stdint.h>

// ---------------- problem constants ----------------
#define NB   16          // batch
#define NH   128         // H
#define NW   128         // W
#define NC   80          // C
#define NHW  (NH * NW)   // 16384
#define ROWE (NW * NC)   // 10240 elements per (b,h) row
#define TOPK 100
#define CAND 128         // padded per-row candidate slots
#define ROWS (NB * NH)   // 2048

// =====================================================================
// Kernel 1: per-(b,h) row — async-stage row to LDS, separable 5x5 (W,C)
// window max (-inf padding), peak-suppress, row top-(>=100) selection.
// Thread map: t = 2*w + half; each thread owns 40 contiguous channels.
// =====================================================================
__global__ void __launch_bounds__(256)
cnet_row_suppress_topk(const float* __restrict__ conf, uint2* __restrict__ cand) {
    __shared__ __align__(16) float rowv[ROWE];   // 40 KB staged row (reused for cmax)
    __shared__ int s_count;                      // reduction scratch
    __shared__ int s_cnt;                        // compaction counter

    const int row = blockIdx.x;                  // b*128 + h
    const int t   = threadIdx.x;
    const int w   = t >> 1;                      // 0..127
    const int c0  = (t & 1) * 40;                // 0 or 40
    const float* __restrict__ src = conf + (size_t)row * ROWE;

    // ---- CDNA5 async global->LDS staging: 2560 B128 transfers ----
    #pragma unroll
    for (int i = 0; i < 10; ++i) {
        const int v4 = t + i * 256;                          // vec4 index
        uint32_t lds_off = (uint32_t)(uintptr_t)(void*)&rowv[v4 * 4];
        uint64_t gaddr   = (uint64_t)(uintptr_t)(const void*)(src + v4 * 4);
        asm volatile("global_load_async_to_lds_b128 %0, %1, off"
                     :: "v"(lds_off), "v"(gaddr) : "memory");
    }
    asm volatile("s_wait_asynccnt 0" ::: "memory");
    __syncthreads();

    // ---- Phase A: own 40 values + 2-wide C halo (same w row) ----
    float a[44];                                 // c = c0-2 .. c0+41
    {
        const float4* p4 = (const float4*)&rowv[w * NC + c0];
        #pragma unroll
        for (int i = 0; i < 10; ++i) {
            const float4 q = p4[i];
            a[2 + 4*i + 0] = q.x; a[2 + 4*i + 1] = q.y;
            a[2 + 4*i + 2] = q.z; a[2 + 4*i + 3] = q.w;
        }
        a[0]  = (c0 >= 2)      ? rowv[w * NC + c0 - 2]  : -1.0f;  // -inf pad
        a[1]  = (c0 >= 1)      ? rowv[w * NC + c0 - 1]  : -1.0f;
        a[42] = (c0 + 40 < NC) ? rowv[w * NC + c0 + 40] : -1.0f;
        a[43] = (c0 + 41 < NC) ? rowv[w * NC + c0 + 41] : -1.0f;
    }
    // 5-wide window max along C, in registers
    float cm[40];
    #pragma unroll
    for (int j = 0; j < 40; ++j)
        cm[j] = fmaxf(fmaxf(a[j], a[j + 1]),
                      fmaxf(a[j + 2], fmaxf(a[j + 3], a[j + 4])));

    __syncthreads();                             // all Phase-A reads done

    // ---- Phase B: overwrite rowv with cmax (in place, no 2nd buffer) ----
    {
        float4* p4 = (float4*)&rowv[w * NC + c0];
        #pragma unroll
        for (int i = 0; i < 10; ++i)
            p4[i] = make_float4(cm[4*i], cm[4*i + 1], cm[4*i + 2], cm[4*i + 3]);
    }
    __syncthreads();

    // ---- Phase C: 5-wide window max along W over cmax rows (b128 reads) ----
    float hm[40];
    #pragma unroll
    for (int j = 0; j < 40; ++j) hm[j] = -1.0f;
    #pragma unroll
    for (int dw = -2; dw <= 2; ++dw) {
        const int ww = w + dw;
        if (ww >= 0 && ww < NW) {
            const float4* p4 = (const float4*)&rowv[ww * NC + c0];
            #pragma unroll
            for (int i = 0; i < 10; ++i) {
                const float4 q = p4[i];
                hm[4*i + 0] = fmaxf(hm[4*i + 0], q.x);
                hm[4*i + 1] = fmaxf(hm[4*i + 1], q.y);
                hm[4*i + 2] = fmaxf(hm[4*i + 2], q.z);
                hm[4*i + 3] = fmaxf(hm[4*i + 3], q.w);
            }
        }
    }
    unsigned keys[40];                           // suppressed score bits (>=0)
    #pragma unroll
    for (int j = 0; j < 40; ++j) {
        const float v = a[2 + j];
        keys[j] = (hm[j] == v) ? __float_as_uint(v) : 0u;
    }

    // block-uniform count of keys >= p; wave32 reduce + 1 atomic per wave
    auto blockCountGE = [&](unsigned p) -> int {
        if (t == 0) s_count = 0;
        __syncthreads();
        int loc = 0;
        #pragma unroll
        for (int j = 0; j < 40; ++j) loc += (keys[j] >= p) ? 1 : 0;
        #pragma unroll
        for (int off = 16; off > 0; off >>= 1) loc += __shfl_xor(loc, off, 32);
        if ((t & 31) == 0) atomicAdd(&s_count, loc);
        __syncthreads();
        const int r = s_count;
        __syncthreads();
        return r;
    };

    // ---- threshold: any T with TOPK <= count(>=T) <= CAND is a valid superset
    unsigned T;
    const int nz = blockCountGE(1u);
    if (nz <= CAND) {
        T = 1u;                                  // take all nonzero peaks
    } else {
        unsigned lo = 1u, hi = 0x3F800000u;      // scores < 1.0f strictly
        T = lo;
        for (int it = 0; it < 32; ++it) {
            const unsigned mid = lo + ((hi - lo) >> 1);
            const int c = blockCountGE(mid);
            if (c >= TOPK && c <= CAND) { T = mid; break; }   // uniform exit
            if (c >= TOPK) lo = mid; else hi = mid;
            T = lo;                              // invariant: count(>=lo) >= TOPK
        }
    }

    // ---- compact (key, hw) pairs, pad to CAND ----
    if (t == 0) s_cnt = 0;
    __syncthreads();
    uint2* __restrict__ dst = cand + (size_t)row * CAND;
    const unsigned hw = (unsigned)((row & (NH - 1)) * NW + w);  // same for all 40
    #pragma unroll
    for (int j = 0; j < 40; ++j) {
        if (keys[j] >= T) {
            const int p = atomicAdd(&s_cnt, 1);
            if (p < CAND) dst[p] = make_uint2(keys[j], hw);
        }
    }
    __syncthreads();
    for (int p = min(s_cnt, CAND) + t; p < CAND; p += 256)
        dst[p] = make_uint2(0u, 0u);
}

// =====================================================================
// Kernel 2: per-batch merge of 128x128 row candidates -> sorted top-100.
// =====================================================================
__global__ void __launch_bounds__(256)
cnet_batch_topk(const uint2* __restrict__ cand, uint2* __restrict__ sel) {
    const int b = blockIdx.x;
    const int t = threadIdx.x;
    const uint2* __restrict__ src = cand + (size_t)b * NH * CAND;  // 16384 entries
    __shared__ int  s_count;
    __shared__ int  s_cnt;
    __shared__ uint2 s_sel[256];

    auto countGE = [&](unsigned p) -> int {
        if (t == 0) s_count = 0;
        __syncthreads();
        int loc = 0;
        for (int i = t; i < NH * CAND; i += 256) loc += (src[i].x >= p) ? 1 : 0;
        #pragma unroll
        for (int off = 16; off > 0; off >>= 1) loc += __shfl_xor(loc, off, 32);
        if ((t & 31) == 0) atomicAdd(&s_count, loc);
        __syncthreads();
        const int r = s_count;
        __syncthreads();
        return r;
    };

    unsigned T;
    const int nz = countGE(1u);
    if (nz <= TOPK) {
        T = 1u;
    } else {
        unsigned lo = 1u, hi = 0x3F800000u;
        T = lo;
        for (int it = 0; it < 32; ++it) {
            const unsigned mid = lo + ((hi - lo) >> 1);
            const int c = countGE(mid);
            if (c >= TOPK && c <= 256) { T = mid; break; }
            if (c >= TOPK) lo = mid; else hi = mid;
            T = lo;
        }
    }

    // collect survivors (typically ~100, cap 256)
    if (t == 0) s_cnt = 0;
    __syncthreads();
    for (int i = t; i < NH * CAND; i += 256) {
        const uint2 e = src[i];
        if (e.x >= T) {
            const int p = atomicAdd(&s_cnt, 1);
            if (p < 256) s_sel[p] = e;
        }
    }
    __syncthreads();
    const int n = min(s_cnt, 256);

    // O(n^2) rank sort, descending by key; ranks form a dense permutation
    uint2* __restrict__ out = sel + (size_t)b * TOPK;
    if (t < n) {
        const uint2 me = s_sel[t];
        int r = 0;
        for (int j = 0; j < n; ++j) {
            const uint2 o = s_sel[j];
            if (o.x > me.x || (o.x == me.x && j < t)) ++r;
        }
        if (r < TOPK) out[r] = me;
    }
    for (int k = min(n, TOPK) + t; k < TOPK; k += 256)
        out[k] = make_uint2(0u, 0u);
}

// =====================================================================
// Kernel 3: gather output row [boxes*4 | 1 | suppressed class row(80)].
// =====================================================================
__global__ void __launch_bounds__(128)
cnet_gather(const float* __restrict__ boxes, const float* __restrict__ conf,
            const uint2* __restrict__ sel, float* __restrict__ out) {
    const int bk = blockIdx.x;               // b*100 + k
    const int b  = bk / TOPK;
    const uint2 s = sel[bk];
    const int hw = (int)s.y;
    const int h  = hw >> 7;
    const int w  = hw & (NW - 1);
    float* __restrict__ o = out + (size_t)bk * (5 + NC);
    const float* __restrict__ rowc = conf + ((size_t)b * NH + h) * (size_t)ROWE;
    const int t = threadIdx.x;

    if (t < NC) {
        const int c = t;
        const float v = rowc[w * NC + c];
        const int w0 = max(w - 2, 0), w1 = min(w + 2, NW - 1);
        const int c0 = max(c - 2, 0), c1 = min(c + 2, NC - 1);
        float m = -1.0f;
        for (int ww = w0; ww <= w1; ++ww)
            for (int cc = c0; cc <= c1; ++cc)
                m = fmaxf(m, rowc[ww * NC + cc]);
        o[5 + c] = (m == v) ? v : 0.0f;
    } else if (t < NC + 4) {
        const int j = t - NC;
        o[j] = boxes[((size_t)b * NHW + hw) * 4 + j] * 4.0f;   // STRIDE = 4.0
    } else if (t == NC + 4) {
        o[4] = 1.0f;
    }
}

// =====================================================================
extern "C" void kernel_launch(void* const* d_in, const int* in_sizes, int n_in,
                              void* d_out, int out_size, void* d_ws, size_t ws_size,
                              hipStream_t stream) {
    (void)in_sizes; (void)n_in; (void)out_size; (void)ws_size;
    const float* boxes = (const float*)d_in[0];   // [16,128,128,4]
    const float* conf  = (const float*)d_in[1];   // [16,128,128,80]
    float* out = (float*)d_out;                   // [16,100,85]

    uint2* cand = (uint2*)d_ws;                   // 2048 * 128 entries (2 MB)
    uint2* sel  = cand + (size_t)ROWS * CAND;     // 16 * 100 entries

    cnet_row_suppress_topk<<<ROWS, 256, 0, stream>>>(conf, cand);
    cnet_batch_topk<<<NB, 256, 0, stream>>>(cand, sel);
    cnet_gather<<<NB * TOPK, 128, 0, stream>>>(boxes, conf, sel, out);
}